// YOLOV1Loss_53944789238331
// MI455X (gfx1250) — compile-verified
//
#include <hip/hip_runtime.h>
#include <hip/hip_bf16.h>

// ---------------------------------------------------------------------------
// YOLOv1 loss, MI455X (gfx1250, wave32).
// Memory-bound: predicate heavy loads on tobj>0 (~70% of cells skip 208B),
// float2 loads where 8B alignment is provable, global_prefetch_b8 streaming,
// and an exact f32 WMMA (V_WMMA_F32_16X16X4_F32, B=ones) wave reduction.
// ---------------------------------------------------------------------------

#define S_DIM 7
#define B_DIM 2
#define C_DIM 20
#define SS 49                    // S*S
#define PRED_STRIDE 1470         // SS*(C + B + 4B) = 49*30
#define TGT_STRIDE  1225         // SS*(C + 1 + 4)  = 49*25
#define PO_OFF      980          // SS*C
#define PCOORD_OFF  1078         // SS*(C+B)
#define TOBJ_OFF    980          // SS*C
#define TCOORD_OFF  1029         // SS*(C+1)
#define COORD_SCALE 5.0f
#define NOOBJ_SCALE 0.5f

typedef __attribute__((ext_vector_type(2))) float v2f;
typedef __attribute__((ext_vector_type(8))) float v8f;

// Exact full-precision wave32 reduction using the f32 WMMA pipe.
// A(16x4): a0 = acc (K=0 lanes0-15 / K=2 lanes16-31), a1 = 0.
// B(4x16) = all ones.  D[m][n] = acc(m) + acc(m+16).
// D layout: d[v] at lane L holds rowsum(M = v + (L>=16 ? 8 : 0)).
// sum(d[0..7]) at lane0 = acc(0..7)+acc(16..23); at lane16 = the rest.
// One shfl_xor(16) completes the reduction. Requires EXEC == all ones.
__device__ __forceinline__ float wave_reduce_wmma(float acc) {
    v2f a; a[0] = acc;  a[1] = 0.0f;
    v2f b; b[0] = 1.0f; b[1] = 1.0f;
    v8f c = {};
    v8f d = __builtin_amdgcn_wmma_f32_16x16x4_f32(
        /*neg_a=*/false, a, /*neg_b=*/false, b,
        /*c_mod=*/(short)0, c, /*reuse_a=*/false, /*reuse_b=*/false);
    float s = ((d[0] + d[1]) + (d[2] + d[3])) + ((d[4] + d[5]) + (d[6] + d[7]));
    s += __shfl_xor(s, 16, 32);
    return s;   // valid at lanes 0 and 16
}

__global__ void __launch_bounds__(256)
yolov1_loss_partial(const float* __restrict__ preds,
                    const float* __restrict__ targets,
                    float* __restrict__ partial,
                    int total_cells) {
    const int tid    = blockIdx.x * blockDim.x + threadIdx.x;
    const int stride = gridDim.x * blockDim.x;

    float acc = 0.0f;

    for (int i = tid; i < total_cells; i += stride) {
        const int n = i / SS;
        const int s = i - n * SS;
        const float* __restrict__ pr = preds   + (size_t)n * PRED_STRIDE;
        const float* __restrict__ tg = targets + (size_t)n * TGT_STRIDE;

        // Prefetch next iteration's rows (streaming, one pass over 353MB).
        {
            int j = i + stride;
            if (j < total_cells) {
                int nn = j / SS;
                __builtin_prefetch(preds   + (size_t)nn * PRED_STRIDE, 0, 0);
                __builtin_prefetch(targets + (size_t)nn * TGT_STRIDE, 0, 0);
            }
        }

        const float tobj = tg[TOBJ_OFF + s];

        // po offset is provably even -> float2 load is 8B aligned.
        const float2 po = *reinterpret_cast<const float2*>(pr + PO_OFF + s * B_DIM);
        const float po0 = po.x, po1 = po.y;

        float cell_loss;
        if (tobj > 0.0f) {
            // ---- class loss: sum_c (pc - tc)^2  (pc 8B aligned, tc not) ----
            const float2* __restrict__ pc2 =
                reinterpret_cast<const float2*>(pr + s * C_DIM);
            const float* __restrict__ tc = tg + s * C_DIM;
            float lc = 0.0f;
            #pragma unroll
            for (int k = 0; k < C_DIM / 2; ++k) {
                float2 p = pc2[k];
                float d0 = p.x - tc[2 * k];
                float d1 = p.y - tc[2 * k + 1];
                lc = fmaf(d0, d0, lc);
                lc = fmaf(d1, d1, lc);
            }

            // ---- boxes ----
            const float2* __restrict__ pco2 =
                reinterpret_cast<const float2*>(pr + PCOORD_OFF + s * (B_DIM * 4));
            const float* __restrict__ tco = tg + TCOORD_OFF + s * 4;

            const float tx = tco[0] * (1.0f / (float)S_DIM);
            const float ty = tco[1] * (1.0f / (float)S_DIM);
            const float tw = tco[2];
            const float th = tco[3];
            const float t0 = tx - 0.5f * tw, t1 = ty - 0.5f * th;
            const float t2 = tx + 0.5f * tw, t3 = ty + 0.5f * th;
            const float area_t = tw * th;

            float iou[B_DIM], rmse2[B_DIM];
            float bxv[B_DIM], byv[B_DIM], bwv[B_DIM], bhv[B_DIM];
            #pragma unroll
            for (int bb = 0; bb < B_DIM; ++bb) {
                float2 lo = pco2[bb * 2 + 0];   // x, y
                float2 hi = pco2[bb * 2 + 1];   // w, h
                bxv[bb] = lo.x; byv[bb] = lo.y; bwv[bb] = hi.x; bhv[bb] = hi.y;
                const float px  = lo.x * (1.0f / (float)S_DIM);
                const float py  = lo.y * (1.0f / (float)S_DIM);
                const float pw2 = hi.x * hi.x;
                const float ph2 = hi.y * hi.y;
                const float p0 = px - 0.5f * pw2, p1 = py - 0.5f * ph2;
                const float p2 = px + 0.5f * pw2, p3 = py + 0.5f * ph2;
                const float iw = fmaxf(fminf(p2, t2) - fmaxf(p0, t0), 0.0f);
                const float ih = fmaxf(fminf(p3, t3) - fmaxf(p1, t1), 0.0f);
                const float inter  = iw * ih;
                const float area_p = pw2 * ph2;
                iou[bb] = inter / (area_p + area_t - inter);   // denom > 0 always
                const float d0 = p0 - t0, d1 = p1 - t1;
                const float d2 = p2 - t2, d3 = p3 - t3;
                rmse2[bb] = d0 * d0 + d1 * d1 + d2 * d2 + d3 * d3; // sqrt monotone
            }

            const float max_iou  = fmaxf(iou[0], iou[1]);
            const int   iou_idx  = (iou[1]   > iou[0])   ? 1 : 0;  // argmax, first
            const int   rmse_idx = (rmse2[1] < rmse2[0]) ? 1 : 0;  // argmin, first
            const int   best     = (max_iou == 0.0f) ? rmse_idx : iou_idx;

            const float bx = best ? bxv[1] : bxv[0];
            const float by = best ? byv[1] : byv[0];
            const float bw = best ? bwv[1] : bwv[0];
            const float bh = best ? bhv[1] : bhv[0];

            const float dx  = bx - tco[0];
            const float dy  = by - tco[1];
            const float dw  = bw - sqrtf(tw);
            const float dh  = bh - sqrtf(th);
            const float loss_xywh = dx * dx + dy * dy + dw * dw + dh * dh;

            const float bop = best ? po1 : po0;
            const float wop = best ? po0 : po1;
            const float dob = bop - max_iou;

            cell_loss = COORD_SCALE * loss_xywh + dob * dob
                      + NOOBJ_SCALE * (wop * wop) + lc;
        } else {
            // no-object cell: 0.5 * (po0^2 + po1^2)
            cell_loss = NOOBJ_SCALE * (po0 * po0 + po1 * po1);
        }
        acc += cell_loss;
    }

    // ---- wave reduction via f32 WMMA (exact), then LDS block reduction ----
    const float wsum = wave_reduce_wmma(acc);

    __shared__ float smem[8];
    if ((threadIdx.x & 31) == 0) smem[threadIdx.x >> 5] = wsum;
    __syncthreads();
    if (threadIdx.x == 0) {
        float t = 0.0f;
        #pragma unroll
        for (int k = 0; k < 8; ++k) t += smem[k];
        partial[blockIdx.x] = t;
    }
}

__global__ void __launch_bounds__(256)
yolov1_loss_final(const float* __restrict__ partial, int count,
                  float* __restrict__ out, float inv_n) {
    float acc = 0.0f;
    for (int i = threadIdx.x; i < count; i += 256) acc += partial[i];

    const float wsum = wave_reduce_wmma(acc);

    __shared__ float smem[8];
    if ((threadIdx.x & 31) == 0) smem[threadIdx.x >> 5] = wsum;
    __syncthreads();
    if (threadIdx.x == 0) {
        float t = 0.0f;
        #pragma unroll
        for (int k = 0; k < 8; ++k) t += smem[k];
        out[0] = t * inv_n;
    }
}

extern "C" void kernel_launch(void* const* d_in, const int* in_sizes, int n_in,
                              void* d_out, int out_size, void* d_ws, size_t ws_size,
                              hipStream_t stream) {
    const float* preds   = (const float*)d_in[0];
    const float* targets = (const float*)d_in[1];
    float* out = (float*)d_out;
    float* ws  = (float*)d_ws;

    const int N = in_sizes[0] / PRED_STRIDE;
    const int total_cells = N * SS;

    int blocks = 1024;
    // stay within scratch if the harness gave us little
    if (ws_size < (size_t)blocks * sizeof(float)) {
        blocks = (int)(ws_size / sizeof(float));
        if (blocks < 1) blocks = 1;
    }

    yolov1_loss_partial<<<blocks, 256, 0, stream>>>(preds, targets, ws, total_cells);
    yolov1_loss_final<<<1, 256, 0, stream>>>(ws, blocks, out, 1.0f / (float)N);
}